// PDE_pre_71133248357060
// MI455X (gfx1250) — compile-verified
//
#include <hip/hip_runtime.h>

// ---------------------------------------------------------------------------
// Burgers-like stencil, (16,2,1024,1024) f32 -> same shape.
//   out_c = nu[b]*lap(x_c) - (u+v)*grad(x_c)
//   grad = (2c - up - left)/DX ;  lap = (4c - up - left + right - down)/DX^2
// Memory-bound: 256 MB mandatory traffic -> ~11 us @ 23.3 TB/s floor.
// To actually hit that, every wave must move >=~15 B per issue slot, so this
// version is built around wide ops: async b128 global->LDS tile staging
// (ASYNCcnt path), float4 LDS reads, b128 global stores, zero integer
// divisions, static per-wave task mapping.
// ---------------------------------------------------------------------------

#define TW       128
#define TH       16
#define LDS_W    136                   // row stride (floats); interior at col 4 (16B aligned)
#define LDS_ROWS (TH + 2)              // 18
#define LDS_CH   (LDS_ROWS * LDS_W)    // 2448 floats per channel

#if defined(__has_builtin)
#  if __has_builtin(__builtin_amdgcn_global_load_async_to_lds_b32)
#    define HAVE_ASYNC32 1
#  endif
#  if __has_builtin(__builtin_amdgcn_global_load_async_to_lds_b128)
#    define HAVE_ASYNC128 1
#  endif
#endif
#ifndef HAVE_ASYNC32
#  define HAVE_ASYNC32 0
#endif
#ifndef HAVE_ASYNC128
#  define HAVE_ASYNC128 0
#endif

#if defined(__AMDGCN__) && !HAVE_ASYNC32
#  warning "async b32 builtin unavailable"
#endif
#if defined(__AMDGCN__) && !HAVE_ASYNC128
#  warning "async b128 builtin unavailable"
#endif

typedef __attribute__((ext_vector_type(4))) int int4v;
typedef __attribute__((address_space(1))) int   g_i32;
typedef __attribute__((address_space(3))) int   lds_i32;
typedef __attribute__((address_space(1))) int4v g_v4i;
typedef __attribute__((address_space(3))) int4v lds_v4i;

__global__ __launch_bounds__(256)
void pde_burgers_stencil(const float* __restrict__ x,
                         const float* __restrict__ nu,
                         float* __restrict__ out)
{
    __shared__ float tile[2 * LDS_CH];

    const int tid  = threadIdx.x;
    const int lane = tid & 31;
    const int wv   = tid >> 5;                 // wave id 0..7
    const int b    = blockIdx.z;
    const int h0   = blockIdx.y * TH;
    const int w0   = blockIdx.x * TW;

    const int   chan_stride = 1024 * 1024;     // floats per channel image
    const float nub = nu[b];
    const float* xb = x + b * 2 * chan_stride; // batch base

    // ---- cooperative tile load: 36 row-tasks (2 ch x 18 rows) -------------
    for (int rt = wv; rt < 2 * LDS_ROWS; rt += 8) {
        const int ch = (rt >= LDS_ROWS) ? 1 : 0;
        const int ly = rt - (ch ? LDS_ROWS : 0);
        const int gh = h0 - 1 + ly;
        float* lrow = &tile[ch * LDS_CH + ly * LDS_W];

        if ((unsigned)gh < 1024u) {
            const float* grow = xb + ch * chan_stride + (gh << 10) + w0;
            // interior: 128 floats = 4 per lane, one async b128 per wave-row
#if HAVE_ASYNC128
            __builtin_amdgcn_global_load_async_to_lds_b128(
                (g_v4i*)(grow + 4 * lane), (lds_v4i*)(lrow + 4 + 4 * lane), 0, 0);
#else
            *(float4*)(lrow + 4 + 4 * lane) = *(const float4*)(grow + 4 * lane);
#endif
            // halo columns: lane 0 -> left (col 3), lane 1 -> right (col 132)
            if (lane < 2) {
                const int hcol = lane ? (4 + TW) : 3;
                const int gwh  = lane ? (w0 + TW) : (w0 - 1);
                if ((unsigned)gwh < 1024u) {
#if HAVE_ASYNC32
                    __builtin_amdgcn_global_load_async_to_lds_b32(
                        (g_i32*)(xb + ch * chan_stride + (gh << 10) + gwh),
                        (lds_i32*)(lrow + hcol), 0, 0);
#else
                    lrow[hcol] = xb[ch * chan_stride + (gh << 10) + gwh];
#endif
                } else {
                    lrow[hcol] = 0.0f;         // zero pad (no async racing this addr)
                }
            }
        } else {
            // whole row out of range: zero cols 0..135
            const float4 z = make_float4(0.f, 0.f, 0.f, 0.f);
            *(float4*)(lrow + 4 * lane) = z;                    // cols 0..127
            if (lane < 2) *(float4*)(lrow + 128 + 4 * lane) = z; // cols 128..135
        }
    }

#if defined(__AMDGCN__) && defined(__has_builtin)
#  if __has_builtin(__builtin_amdgcn_s_wait_asynccnt)
    __builtin_amdgcn_s_wait_asynccnt(0);
#  endif
#endif
    __syncthreads();

    // ---- compute: thread owns 4 consecutive w, rows rq and rq+8, both ch --
    const float invdx  = 2.54647908947032526f;   // 16 / (2*pi)
    const float invdx2 = invdx * invdx;
    const float ns2    = nub * invdx2;

    const int wq = tid & 31;     // w chunk: cols 4*wq .. 4*wq+3
    const int rq = tid >> 5;     // 0..7

    const float* lu = tile;
    const float* lv = tile + LDS_CH;
    const int    cc = 4 + 4 * wq;

    auto stencil2 = [&](float cU, float upU, float lU, float rU, float dU,
                        float cV, float upV, float lV, float rV, float dV,
                        float& oU, float& oV) {
        const float tU   = upU + lU;
        const float aU   = 2.0f * cU - tU;              // grad numerator
        const float lapU = 4.0f * cU - tU + rU - dU;    // lap numerator
        const float tV   = upV + lV;
        const float aV   = 2.0f * cV - tV;
        const float lapV = 4.0f * cV - tV + rV - dV;
        const float si   = (cU + cV) * invdx;           // (u+v)/DX, shared
        oU = ns2 * lapU - si * aU;
        oV = ns2 * lapV - si * aV;
    };

#pragma unroll
    for (int rr = 0; rr < 2; ++rr) {
        const int r  = rq + rr * 8;                    // output row 0..15
        const int rb = (r + 1) * LDS_W + cc;           // center LDS index

        const float4 uc = *(const float4*)&lu[rb];
        const float4 uu = *(const float4*)&lu[rb - LDS_W];
        const float4 ud = *(const float4*)&lu[rb + LDS_W];
        const float  ulh = lu[rb - 1], urh = lu[rb + 4];

        const float4 vc = *(const float4*)&lv[rb];
        const float4 vu = *(const float4*)&lv[rb - LDS_W];
        const float4 vd = *(const float4*)&lv[rb + LDS_W];
        const float  vlh = lv[rb - 1], vrh = lv[rb + 4];

        float4 o0, o1;
        stencil2(uc.x, uu.x, ulh,  uc.y, ud.x,  vc.x, vu.x, vlh,  vc.y, vd.x, o0.x, o1.x);
        stencil2(uc.y, uu.y, uc.x, uc.z, ud.y,  vc.y, vu.y, vc.x, vc.z, vd.y, o0.y, o1.y);
        stencil2(uc.z, uu.z, uc.y, uc.w, ud.z,  vc.z, vu.z, vc.y, vc.w, vd.z, o0.z, o1.z);
        stencil2(uc.w, uu.w, uc.z, urh,  ud.w,  vc.w, vu.w, vc.z, vrh,  vd.w, o0.w, o1.w);

        float* op = out + ((b * 2) * 1024 + (h0 + r)) * 1024 + (w0 + 4 * wq);
        *(float4*)op                 = o0;
        *(float4*)(op + chan_stride) = o1;
    }
}

extern "C" void kernel_launch(void* const* d_in, const int* in_sizes, int n_in,
                              void* d_out, int out_size, void* d_ws, size_t ws_size,
                              hipStream_t stream) {
    (void)in_sizes; (void)n_in; (void)out_size; (void)d_ws; (void)ws_size;
    const float* x  = (const float*)d_in[0];
    const float* nu = (const float*)d_in[1];
    float* out      = (float*)d_out;

    dim3 grid(1024 / TW, 1024 / TH, 16);   // 8 x 64 x 16 = 8192 blocks
    dim3 block(256);                       // 8 wave32s
    hipLaunchKernelGGL(pde_burgers_stencil, grid, block, 0, stream, x, nu, out);
}